// LightGCN_52458730553910
// MI455X (gfx1250) — compile-verified
//
#include <hip/hip_runtime.h>
#include <math.h>

#define EMB_DIM 128
#define N_USERS 100000

typedef __attribute__((ext_vector_type(2))) float v2f;
typedef __attribute__((ext_vector_type(8))) float v8f;

// Hardware FP32 atomic add (no return): single global_atomic_add_f32 to the L2
// atomic units, tracked with STOREcnt; S_ENDPGM's implicit wait-idle + stream
// ordering make results visible to subsequent kernels. Avoids any chance of a
// CAS-loop lowering for the ~230M edge-scatter atomics.
__device__ __forceinline__ void atomAddF(float* p, float v) {
  asm volatile("global_atomic_add_f32 %0, %1, off" : : "v"(p), "v"(v) : "memory");
}

// ---------------- zero fills ----------------
__global__ void zero_f32(float* __restrict__ p, long n) {
  long i = (long)blockIdx.x * blockDim.x + threadIdx.x;
  if (i < n) p[i] = 0.0f;
}

__global__ void zero_f32x4(float4* __restrict__ p, long n4) {
  long i = (long)blockIdx.x * blockDim.x + threadIdx.x;
  if (i < n4) p[i] = make_float4(0.f, 0.f, 0.f, 0.f);
}

// ---------------- degree + normalization ----------------
__global__ void count_deg(const int* __restrict__ rows, float* __restrict__ deg, int nE) {
  int e = blockIdx.x * blockDim.x + threadIdx.x;
  if (e < nE) atomAddF(&deg[rows[e]], 1.0f);
}

__global__ void compute_dinv(const float* __restrict__ deg, float* __restrict__ dinv, long n) {
  long i = (long)blockIdx.x * blockDim.x + threadIdx.x;
  if (i < n) {
    float d = deg[i];
    dinv[i] = (d > 0.0f) ? rsqrtf(fmaxf(d, 1.0f)) : 0.0f;
  }
}

// ---------------- edge scatter: out[col] += in[row] * dinv[row]^pw ----------------
// One wave32 per edge; lane l covers dims [4l, 4l+4). Normalization of the
// *previous* layer's destination (dc) is folded into this read as dinv^2,
// eliminating the full elementwise x=agg*dc pass over the 76.8MB table.
// The 150K x 128 fp32 node table (76.8MB) is resident in the 192MB L2, so the
// random row gathers and the atomic accumulates are L2 traffic, not HBM.
__global__ void scatter_edges(const float* __restrict__ in,
                              const int* __restrict__ rows,
                              const int* __restrict__ cols,
                              const float* __restrict__ dinv,
                              float* __restrict__ out, int nE, int pw) {
  int wid  = (int)(((long)blockIdx.x * blockDim.x + threadIdx.x) >> 5);
  int lane = threadIdx.x & 31;
  if (wid >= nE) return;
  int r = rows[wid];
  int c = cols[wid];
  float s = dinv[r];
  if (pw == 2) s *= s;
  float4 v = *(const float4*)(in + (size_t)r * EMB_DIM + lane * 4);
  float* ob = out + (size_t)c * EMB_DIM + lane * 4;
  atomAddF(ob + 0, v.x * s);
  atomAddF(ob + 1, v.y * s);
  atomAddF(ob + 2, v.z * s);
  atomAddF(ob + 3, v.w * s);
}

// ---------------- gather the 3*BATCH rows the loss needs, accumulate ----------------
// mode 0: gacc = emb[node]             (layer-0 term of acc)
// mode 1: gacc += raw[node]*dinv[node] (x_l[node] recovered from unscaled agg)
__global__ void gather_acc(const float* __restrict__ src,
                           const int* __restrict__ users,
                           const int* __restrict__ pos,
                           const int* __restrict__ neg,
                           const float* __restrict__ dinv,
                           float* __restrict__ gacc, int batch, int mode) {
  int wid  = (int)(((long)blockIdx.x * blockDim.x + threadIdx.x) >> 5);
  int lane = threadIdx.x & 31;
  int total = 3 * batch;
  if (wid >= total) return;
  int node;
  if (wid < batch)            node = users[wid];
  else if (wid < 2 * batch)   node = N_USERS + pos[wid - batch];
  else                        node = N_USERS + neg[wid - 2 * batch];
  float4 v = *(const float4*)(src + (size_t)node * EMB_DIM + lane * 4);
  float4* g = (float4*)(gacc + (size_t)wid * EMB_DIM + lane * 4);
  if (mode == 0) {
    *g = v;
  } else {
    float s = dinv[node];
    float4 o = *g;
    o.x += v.x * s; o.y += v.y * s; o.z += v.z * s; o.w += v.w * s;
    *g = o;
  }
}

// ---------------- WMMA scoring + BPR loss ----------------
// One wave per 16-triple tile. C = U x P^T via V_WMMA_F32_16X16X4_F32 over 32
// K-chunks; the diagonal holds the 16 dot products. The /4 on acc appears
// squared in the score, folded as 1/16 into the score difference.
// f32 16x16x4 operand layout: lane<16 -> m=lane,k in {0,1}; lane>=16 -> m=lane-16,k in {2,3}.
// B (4x16, K x N) mirrors this with N in place of M, so B = P^T loads with the
// identical per-lane indexing applied to P.
__global__ void score_loss(const float* __restrict__ gacc, float* __restrict__ out, int batch) {
  int tile = blockIdx.x;
  int lane = threadIdx.x;      // blockDim.x == 32 (one wave, EXEC all ones)
  int m    = lane & 15;
  int kb   = (lane >> 4) * 2;

  const float* U = gacc + (size_t)(tile * 16) * EMB_DIM;
  const float* P = gacc + (size_t)(batch + tile * 16) * EMB_DIM;
  const float* N = gacc + (size_t)(2 * batch + tile * 16) * EMB_DIM;

  v8f cp = {};
  v8f cn = {};
  const size_t rowOff = (size_t)m * EMB_DIM;
#pragma unroll 4
  for (int k0 = 0; k0 < EMB_DIM; k0 += 4) {
    v2f a, bp, bn;
    a.x  = U[rowOff + k0 + kb];
    a.y  = U[rowOff + k0 + kb + 1];
    bp.x = P[rowOff + k0 + kb];
    bp.y = P[rowOff + k0 + kb + 1];
    bn.x = N[rowOff + k0 + kb];
    bn.y = N[rowOff + k0 + kb + 1];
    cp = __builtin_amdgcn_wmma_f32_16x16x4_f32(false, a, false, bp, (short)0, cp, false, false);
    cn = __builtin_amdgcn_wmma_f32_16x16x4_f32(false, a, false, bn, (short)0, cn, false, false);
  }

  // Diagonal extraction per C/D layout:
  //   i in [0,8):  lane=i,    vgpr=i
  //   i in [8,16): lane=i+16, vgpr=i-8
  bool valid = (lane < 8) || (lane >= 24);
  int  di    = (lane < 8) ? lane : (lane - 24);
  float ps = cp[di];
  float ns = cn[di];

  float loss = 0.0f;
  if (valid) {
    float d = (ps - ns) * (1.0f / 16.0f);   // (acc/4)*(acc/4) scale
    float x = -d;                            // -log_sigmoid(d) = softplus(-d)
    loss = fmaxf(x, 0.0f) + log1pf(expf(-fabsf(x)));
  }
#pragma unroll
  for (int off = 16; off >= 1; off >>= 1) loss += __shfl_xor(loss, off, 32);
  if (lane == 0) atomAddF(out, loss / (float)batch);
}

extern "C" void kernel_launch(void* const* d_in, const int* in_sizes, int n_in,
                              void* d_out, int out_size, void* d_ws, size_t ws_size,
                              hipStream_t stream) {
  const float* emb   = (const float*)d_in[0];
  const int*   users = (const int*)d_in[1];
  const int*   pos   = (const int*)d_in[2];
  const int*   neg   = (const int*)d_in[3];
  const int*   eidx  = (const int*)d_in[4];

  const int  batch  = in_sizes[1];
  const int  nE     = in_sizes[4] / 2;
  const long nNodes = (long)in_sizes[0] / EMB_DIM;
  const int* rows = eidx;
  const int* cols = eidx + nE;

  // workspace layout (floats): deg | dinv | rawA | rawB | gacc  (~161 MB)
  float* deg  = (float*)d_ws;
  float* dinv = deg  + nNodes;
  float* rawA = dinv + nNodes;
  float* rawB = rawA + nNodes * EMB_DIM;
  float* gacc = rawB + nNodes * EMB_DIM;

  const int TB = 256;
  const long nodeElems = nNodes * EMB_DIM;

  // degrees + normalization
  zero_f32<<<(int)((nNodes + TB - 1) / TB), TB, 0, stream>>>(deg, nNodes);
  count_deg<<<(nE + TB - 1) / TB, TB, 0, stream>>>(rows, deg, nE);
  compute_dinv<<<(int)((nNodes + TB - 1) / TB), TB, 0, stream>>>(deg, dinv, nNodes);

  // acc layer-0 term at the rows the loss needs
  int gatherBlocks = (3 * batch * 32 + TB - 1) / TB;
  gather_acc<<<gatherBlocks, TB, 0, stream>>>(emb, users, pos, neg, dinv, gacc, batch, 0);

  // 3 propagation layers, ping-pong raw buffers, dc folded into next read
  int scatterBlocks = (int)(((long)nE * 32 + TB - 1) / TB);
  long n4 = nodeElems / 4;
  int zeroBlocks = (int)((n4 + TB - 1) / TB);

  for (int l = 0; l < 3; ++l) {
    const float* in = (l == 0) ? emb : ((l == 1) ? rawA : rawB);
    float* out = (l == 1) ? rawB : rawA;   // l0->A, l1->B, l2->A
    int pw = (l == 0) ? 1 : 2;
    zero_f32x4<<<zeroBlocks, TB, 0, stream>>>((float4*)out, n4);
    scatter_edges<<<scatterBlocks, TB, 0, stream>>>(in, rows, cols, dinv, out, nE, pw);
    gather_acc<<<gatherBlocks, TB, 0, stream>>>(out, users, pos, neg, dinv, gacc, batch, 1);
  }

  // scoring + loss (WMMA), d_out zeroed for deterministic atomic accumulate
  zero_f32<<<1, 1, 0, stream>>>((float*)d_out, 1);
  score_loss<<<batch / 16, 32, 0, stream>>>(gacc, (float*)d_out, batch);
}